// MCRGANloss_72524817761045
// MI455X (gfx1250) — compile-verified
//
#include <hip/hip_runtime.h>
#include <hip/hip_bf16.h>
#include <math.h>

#define NCLS 10
#define D    1024
#define D2   ((size_t)D * (size_t)D)
#define TK   32
#define BM   128   // block tile rows (A)
#define BN   64    // block tile cols (B)
#define LDA  132   // padded LDS row stride (floats), 528B = 33*16B
#define LDB  68    // padded LDS row stride (floats), 272B = 17*16B

typedef __attribute__((ext_vector_type(2))) float v2f;
typedef __attribute__((ext_vector_type(8))) float v8f;

// ---------------- counting sort (deterministic, no global atomics) ----------

__global__ __launch_bounds__(256) void k_blockhist(const int* __restrict__ lbl,
                                                   int n, int* __restrict__ bh) {
    __shared__ int h[NCLS];
    int t = threadIdx.x;
    if (t < NCLS) h[t] = 0;
    __syncthreads();
    int i = blockIdx.x * 256 + t;
    if (i < n) atomicAdd(&h[lbl[i]], 1);
    __syncthreads();
    if (t < NCLS) bh[blockIdx.x * NCLS + t] = h[t];
}

__global__ void k_scan(int* __restrict__ bh, int nblk,
                       int* __restrict__ counts, int* __restrict__ offs,
                       float* __restrict__ countsF) {
    if (threadIdx.x != 0 || blockIdx.x != 0) return;
    int c[NCLS];
    for (int j = 0; j < NCLS; ++j) {
        int s = 0;
        for (int b = 0; b < nblk; ++b) s += bh[b * NCLS + j];
        c[j] = s; counts[j] = s; countsF[j] = (float)s;
    }
    int run = 0;
    for (int j = 0; j < NCLS; ++j) {
        offs[j] = run;
        int p = run;
        for (int b = 0; b < nblk; ++b) { int t = bh[b * NCLS + j]; bh[b * NCLS + j] = p; p += t; }
        run += c[j];
    }
}

__global__ __launch_bounds__(256) void k_scatter(const int* __restrict__ lbl, int n,
                                                 const int* __restrict__ pref,
                                                 int* __restrict__ rowperm) {
    __shared__ int L[256];
    int t = threadIdx.x;
    int i = blockIdx.x * 256 + t;
    L[t] = (i < n) ? lbl[i] : -1;
    __syncthreads();
    if (i < n) {
        int l = L[t];
        int r = 0;
        for (int u = 0; u < t; ++u) r += (L[u] == l) ? 1 : 0;
        rowperm[pref[blockIdx.x * NCLS + l] + r] = i;
    }
}

// ---------------- async global->LDS helpers ---------------------------------

__device__ __forceinline__ void async_b128(unsigned lds_addr, const float* gptr) {
    unsigned long long ga = (unsigned long long)(uintptr_t)gptr;
    asm volatile("global_load_async_to_lds_b128 %0, %1, off"
                 :: "v"(lds_addr), "v"(ga) : "memory");
}
__device__ __forceinline__ void wait_async0() {
    asm volatile("s_wait_asynccnt 0x0" ::: "memory");
}

// ---------------- per-class Grams via fp32 WMMA -----------------------------
// grid = (D/BN, D/BM, 20); block = 256 threads = 8 waves.
// Each wave computes a 32x32 output tile (2x2 WMMA register blocking).
// K panels are double-buffered in LDS via async-to-LDS copies.

__global__ __launch_bounds__(256) void k_gram(const float* __restrict__ Z,
                                              const float* __restrict__ Zb,
                                              const int* __restrict__ rowperm,
                                              const int* __restrict__ counts,
                                              const int* __restrict__ offs,
                                              float* __restrict__ Gbase) {
    const int j     = blockIdx.z % NCLS;
    const int which = blockIdx.z / NCLS;
    const float* __restrict__ src = which ? Zb : Z;
    float* __restrict__ G = Gbase + (size_t)blockIdx.z * D2;

    const int a0 = blockIdx.y * BM;
    const int b0 = blockIdx.x * BN;
    const int kbeg = offs[j];
    const int kcnt = counts[j];

    __shared__ float ldsA[2][TK][LDA];
    __shared__ float ldsB[2][TK][LDB];

    const int tid  = threadIdx.x;
    const int lane = tid & 31;
    const int wave = tid >> 5;          // 0..7
    const int wy   = wave >> 1;         // 0..3 -> 32-row strip
    const int wx   = wave & 1;          // 0..1 -> 32-col strip

    v8f acc00 = {}, acc01 = {}, acc10 = {}, acc11 = {};

    const int cidx  = lane & 15;        // m (A) / n (B)
    const int khalf = (lane >> 4) * 2;  // 0 or 2 (K pair base)

    const int NB = (kcnt + TK - 1) / TK;

    // ---- panel loader: async copies (in-range) or zero ds_store (tail) ----
    auto load_panels = [&](int kb, int buf) {
        // A panel: 32 rows x 128 cols = 1024 float4 -> 4 issues of 256 lanes
#pragma unroll
        for (int q = 0; q < 4; ++q) {
            int idx = q * 256 + tid;
            int lr  = idx >> 5;            // 0..31
            int c4  = (idx & 31) * 4;      // 0..124
            int k   = kb + lr;
            float* dst = &ldsA[buf][lr][c4];
            if (k < kcnt) {
                int row = rowperm[kbeg + k];
                async_b128((unsigned)(uintptr_t)dst, src + (size_t)row * D + a0 + c4);
            } else {
                *(float4*)dst = make_float4(0.f, 0.f, 0.f, 0.f);
            }
        }
        // B panel: 32 rows x 64 cols = 512 float4 -> 2 issues
#pragma unroll
        for (int q = 0; q < 2; ++q) {
            int idx = q * 256 + tid;
            int lr  = idx >> 4;            // 0..31
            int c4  = (idx & 15) * 4;      // 0..60
            int k   = kb + lr;
            float* dst = &ldsB[buf][lr][c4];
            if (k < kcnt) {
                int row = rowperm[kbeg + k];
                async_b128((unsigned)(uintptr_t)dst, src + (size_t)row * D + b0 + c4);
            } else {
                *(float4*)dst = make_float4(0.f, 0.f, 0.f, 0.f);
            }
        }
    };

    if (NB > 0) load_panels(0, 0);

    for (int b = 0; b < NB; ++b) {
        const int buf = b & 1;
        wait_async0();            // own async copies for panel b complete
        __syncthreads();          // everyone's copies visible; prev reads done
        if (b + 1 < NB) load_panels((b + 1) * TK, buf ^ 1);

        const float (*LA)[LDA] = ldsA[buf];
        const float (*LB)[LDB] = ldsB[buf];
        const int ra = wy * 32 + cidx;
        const int cb = wx * 32 + cidx;
#pragma unroll
        for (int ks = 0; ks < TK; ks += 4) {
            v2f A0, A1, B0, B1;
            A0.x = LA[ks + khalf + 0][ra];      A0.y = LA[ks + khalf + 1][ra];
            A1.x = LA[ks + khalf + 0][ra + 16]; A1.y = LA[ks + khalf + 1][ra + 16];
            B0.x = LB[ks + khalf + 0][cb];      B0.y = LB[ks + khalf + 1][cb];
            B1.x = LB[ks + khalf + 0][cb + 16]; B1.y = LB[ks + khalf + 1][cb + 16];
            acc00 = __builtin_amdgcn_wmma_f32_16x16x4_f32(false, A0, false, B0, (short)0, acc00, false, false);
            acc01 = __builtin_amdgcn_wmma_f32_16x16x4_f32(false, A0, false, B1, (short)0, acc01, false, false);
            acc10 = __builtin_amdgcn_wmma_f32_16x16x4_f32(false, A1, false, B0, (short)0, acc10, false, false);
            acc11 = __builtin_amdgcn_wmma_f32_16x16x4_f32(false, A1, false, B1, (short)0, acc11, false, false);
        }
        __syncthreads();          // done reading buf before it is overwritten
    }

    // C layout: VGPR v -> M = v + 8*(lane>=16), N = lane&15
    const int coln  = lane & 15;
    const int rhalf = (lane >> 4) << 3;
    {
        const int r0 = a0 + wy * 32 + rhalf;
        const int r1 = a0 + wy * 32 + 16 + rhalf;
        const int c0 = b0 + wx * 32 + coln;
        const int c1 = b0 + wx * 32 + 16 + coln;
#pragma unroll
        for (int v = 0; v < 8; ++v) {
            G[(size_t)(r0 + v) * D + c0] = acc00[v];
            G[(size_t)(r0 + v) * D + c1] = acc01[v];
            G[(size_t)(r1 + v) * D + c0] = acc10[v];
            G[(size_t)(r1 + v) * D + c1] = acc11[v];
        }
    }
}

// ---------------- build A = I + c * G (32 matrices) -------------------------

__global__ __launch_bounds__(256) void k_buildA(const float* __restrict__ G,
                                                float* __restrict__ CH,
                                                const float* __restrict__ countsF,
                                                float nf) {
    const int m = blockIdx.y;
    const size_t e = (size_t)blockIdx.x * 256 + threadIdx.x;
    const int r = (int)(e >> 10), c = (int)(e & (D - 1));
    float g, scal;
    if (m < 2) {                                   // full Gram (Z or Z_bar)
        const float* base = G + (size_t)m * NCLS * D2;
        g = 0.f;
        for (int j = 0; j < NCLS; ++j) g += base[(size_t)j * D2 + e];
        scal = 2048.f / nf;                        // d/(n*eps), eps=0.5
    } else if (m < 12) {                           // class Grams of Z
        int j = m - 2;
        g = G[(size_t)j * D2 + e];
        scal = 2048.f / (countsF[j] + 1e-8f);      // d/(trPi*eps)
    } else if (m < 22) {                           // class Grams of Z_bar
        int j = m - 12;
        g = G[(size_t)(NCLS + j) * D2 + e];
        scal = 2048.f / (countsF[j] + 1e-8f);
    } else {                                       // real+fake combined
        int j = m - 22;
        g = G[(size_t)j * D2 + e] + G[(size_t)(NCLS + j) * D2 + e];
        scal = 1024.f / countsF[j];                // d/(2*cnt*eps)
    }
    CH[(size_t)m * D2 + e] = scal * g + ((r == c) ? 1.0f : 0.0f);
}

// ---------------- Cholesky log-det (one block per matrix) -------------------

__global__ __launch_bounds__(1024) void k_chol(float* __restrict__ CH,
                                               float* __restrict__ logdet) {
    float* A = CH + (size_t)blockIdx.x * D2;
    const int tid = threadIdx.x;
    __shared__ float colv[D];
    float lsum = 0.f;
    for (int k = 0; k < D; ++k) {
        __syncthreads();
        const float piv = sqrtf(A[(size_t)k * D + k]);  // read-only; never rewritten
        if (tid == 0) lsum += logf(piv);
        const float inv = 1.0f / piv;
        for (int i = k + 1 + tid; i < D; i += 1024) {
            float v = A[(size_t)i * D + k] * inv;
            A[(size_t)i * D + k] = v;
            colv[i] = v;
        }
        __syncthreads();
        for (int i = k + 1 + tid; i < D; i += 1024) {
            const float lik = colv[i];
            float* rowp = A + (size_t)i * D;
            for (int j2 = k + 1; j2 <= i; ++j2) rowp[j2] -= lik * colv[j2];
        }
    }
    if (tid == 0) logdet[blockIdx.x] = 2.0f * lsum;
}

// ---------------- final scalar combination ----------------------------------

__global__ void k_final(const float* __restrict__ countsF,
                        const float* __restrict__ ld,
                        float* __restrict__ out, float nf) {
    if (threadIdx.x != 0 || blockIdx.x != 0) return;
    float discZ = 0.5f * ld[0], discZb = 0.5f * ld[1];
    float compZ = 0.f, compZb = 0.f, pc = 0.f;
    for (int j = 0; j < NCLS; ++j) {
        float trPi = countsF[j] + 1e-8f;
        compZ  += trPi / (2.0f * nf) * ld[2 + j];
        compZb += trPi / (2.0f * nf) * ld[12 + j];
        float disc = 0.5f * ld[22 + j];
        float comp = trPi / (4.0f * countsF[j]) * (ld[2 + j] + ld[12 + j]);
        pc += comp - disc;                          // -(disc - comp)
    }
    out[0] = (compZ - discZ) + (compZb - discZb) + pc;
}

// ---------------- launch ----------------------------------------------------

extern "C" void kernel_launch(void* const* d_in, const int* in_sizes, int n_in,
                              void* d_out, int out_size, void* d_ws, size_t ws_size,
                              hipStream_t stream) {
    const float* Z   = (const float*)d_in[0];
    const float* Zb  = (const float*)d_in[1];
    const int*   lbl = (const int*)d_in[2];
    const int n = in_sizes[2];
    const float nf = (float)n;

    // workspace layout
    int*   Wi      = (int*)d_ws;
    int*   counts  = Wi + 0;        // 10
    int*   offs    = Wi + 16;       // 10
    int*   bh      = Wi + 32;       // nblk*10
    int*   rowperm = Wi + 2048;     // n
    float* Wf      = (float*)d_ws;
    float* countsF = Wf + 40960;    // 10
    float* logdet  = Wf + 40992;    // 32
    float* G       = Wf + 65536;    // 20 * D2
    float* CH      = G + 20 * D2;   // 32 * D2

    const int nblk = (n + 255) / 256;

    k_blockhist<<<nblk, 256, 0, stream>>>(lbl, n, bh);
    k_scan<<<1, 1, 0, stream>>>(bh, nblk, counts, offs, countsF);
    k_scatter<<<nblk, 256, 0, stream>>>(lbl, n, bh, rowperm);

    dim3 gg(D / BN, D / BM, 2 * NCLS);
    k_gram<<<gg, 256, 0, stream>>>(Z, Zb, rowperm, counts, offs, G);

    dim3 gb((unsigned)(D2 / 256), 32);
    k_buildA<<<gb, 256, 0, stream>>>(G, CH, countsF, nf);

    k_chol<<<32, 1024, 0, stream>>>(CH, logdet);
    k_final<<<1, 32, 0, stream>>>(countsF, logdet, (float*)d_out, nf);
}